// ChamferLoss_5677946765711
// MI455X (gfx1250) — compile-verified
//
#include <hip/hip_runtime.h>
#include <hip/hip_bf16.h>

// Chamfer loss: pred/target [4, 8192, 32] fp32 -> scalar.
// Core = per-batch GEMM pred @ target^T (M=N=8192, K=32) via
// V_WMMA_F32_16X16X4_F32 (native fp32 matrix op, K=4 chained x8),
// fused with row/col min reductions; sqrt deferred past the min.

typedef __attribute__((ext_vector_type(2))) float v2f;
typedef __attribute__((ext_vector_type(8))) float v8f;

#define NPTS 8192
#define DIMS 32
#define INF_BITS 0x7F800000u

// ---------------------------------------------------------------------------
// Kernel 1: init min buffers to +inf, precompute squared norms of every row.
// One thread per point (32768 threads cover both tensors' rows).
// ---------------------------------------------------------------------------
__global__ __launch_bounds__(256) void chamfer_init(
    const float* __restrict__ pred, const float* __restrict__ target,
    unsigned* __restrict__ row_min, unsigned* __restrict__ col_min,
    float* __restrict__ x2, float* __restrict__ y2)
{
    int i = blockIdx.x * blockDim.x + threadIdx.x;   // 0 .. 32767 == b*8192+n
    if (i >= 4 * NPTS) return;
    row_min[i] = INF_BITS;
    col_min[i] = INF_BITS;
    const float4* p = (const float4*)(pred   + (size_t)i * DIMS);
    const float4* t = (const float4*)(target + (size_t)i * DIMS);
    float sx = 0.f, sy = 0.f;
#pragma unroll
    for (int k = 0; k < DIMS / 4; ++k) {
        float4 a = p[k];
        sx += a.x * a.x + a.y * a.y + a.z * a.z + a.w * a.w;
        float4 b = t[k];
        sy += b.x * b.x + b.y * b.y + b.z * b.z + b.w * b.w;
    }
    x2[i] = sx;
    y2[i] = sy;
}

// ---------------------------------------------------------------------------
// Kernel 2: tiled WMMA distance kernel.
// Block = 256 threads = 8 wave32s; block tile = 128 (n) x 128 (m).
// Wave w owns rows [nb*128 + w*16, +16), sweeps 8 column tiles of 16.
//
// fp32 WMMA fragment layouts (CDNA5 ISA 7.12.2):
//   A 16x4:  lane(l) -> M = l&15, VGPR{0,1} -> K = 2*(l>>4) + {0,1}
//   B 4x16:  lane(l) -> N = l&15, VGPR{0,1} -> K = 2*(l>>4) + {0,1}
//   C 16x16: VGPR j, lane(l) -> M = j + 8*(l>>4), N = l&15
// B = target^T, so B loads use the same 8-byte contiguous pattern as A.
// ---------------------------------------------------------------------------
__global__ __launch_bounds__(256) void chamfer_tiles(
    const float* __restrict__ pred, const float* __restrict__ target,
    unsigned* __restrict__ row_min, unsigned* __restrict__ col_min,
    const float* __restrict__ x2, const float* __restrict__ y2)
{
    __shared__ unsigned lds_rowmin[128];
    __shared__ unsigned lds_colmin[128];

    const int tid = threadIdx.x;
    if (tid < 128)       lds_rowmin[tid]       = INF_BITS;
    else if (tid < 256)  lds_colmin[tid - 128] = INF_BITS;
    __syncthreads();

    const int b  = blockIdx.z;      // batch
    const int nb = blockIdx.y;      // 128-row block of pred
    const int mb = blockIdx.x;      // 128-col block of target
    const int w    = tid >> 5;      // wave id 0..7
    const int lane = tid & 31;
    const int lo = lane & 15;
    const int hi = lane >> 4;

    const int n_base  = nb * 128 + w * 16;  // this wave's 16 rows
    const int m_block = mb * 128;

    const float* predB = pred   + (size_t)b * NPTS * DIMS;
    const float* tgtB  = target + (size_t)b * NPTS * DIMS;

    // ---- A fragments (loaded once; 8 x K=4 chunks covering K=32) ----
    v2f a[8];
    const float* arow = predB + (size_t)(n_base + lo) * DIMS + 2 * hi;
#pragma unroll
    for (int f = 0; f < 8; ++f)
        a[f] = *(const v2f*)(arow + 4 * f);

    // ---- squared norms of this lane's 8 output rows (M = j + 8*hi) ----
    float x2v[8];
    const float* x2p = x2 + (size_t)b * NPTS + n_base + 8 * hi;
#pragma unroll
    for (int j = 0; j < 8; ++j) x2v[j] = x2p[j];

    float rmin[8];
#pragma unroll
    for (int j = 0; j < 8; ++j) rmin[j] = __builtin_inff();

    // ---- sweep 8 column tiles ----
#pragma unroll
    for (int t = 0; t < 8; ++t) {
        const int m_base = m_block + t * 16;

        v2f bf[8];
        const float* brow = tgtB + (size_t)(m_base + lo) * DIMS + 2 * hi;
#pragma unroll
        for (int f = 0; f < 8; ++f)
            bf[f] = *(const v2f*)(brow + 4 * f);

        v8f c = {};
#pragma unroll
        for (int f = 0; f < 8; ++f)
            c = __builtin_amdgcn_wmma_f32_16x16x4_f32(
                    /*neg_a=*/false, a[f], /*neg_b=*/false, bf[f],
                    /*c_mod=*/(short)0, c, /*reuse_a=*/false, /*reuse_b=*/false);

        const float y2col = y2[(size_t)b * NPTS + m_base + lo];  // N = lo

        float cmin = __builtin_inff();
#pragma unroll
        for (int j = 0; j < 8; ++j) {
            float d2 = fmaxf(x2v[j] + y2col - 2.0f * c[j], 0.0f);  // >= 0
            rmin[j] = fminf(rmin[j], d2);
            cmin    = fminf(cmin, d2);
        }
        // d2 >= 0  =>  IEEE float order == uint bit order
        atomicMin(&lds_colmin[t * 16 + lo], __float_as_uint(cmin));
    }

    // row mins: register-accumulated over all 8 col tiles, one LDS pass
#pragma unroll
    for (int j = 0; j < 8; ++j)
        atomicMin(&lds_rowmin[w * 16 + 8 * hi + j], __float_as_uint(rmin[j]));

    __syncthreads();

    // one global atomic per output entry per workgroup
    if (tid < 128) {
        atomicMin(&row_min[(size_t)b * NPTS + nb * 128 + tid], lds_rowmin[tid]);
    } else if (tid < 256) {
        const int i = tid - 128;
        atomicMin(&col_min[(size_t)b * NPTS + mb * 128 + i], lds_colmin[i]);
    }
}

// ---------------------------------------------------------------------------
// Kernel 3: final reduction. sqrt(min d2) == min dist (sqrt monotone, d2>=0).
// mean over B*N rows + mean over B*M cols, same denominator (32768).
// ---------------------------------------------------------------------------
__global__ __launch_bounds__(1024) void chamfer_final(
    const unsigned* __restrict__ row_min, const unsigned* __restrict__ col_min,
    float* __restrict__ out)
{
    __shared__ float red[1024];
    const int tid = threadIdx.x;
    const int total = 4 * NPTS;
    float s = 0.f;
    for (int i = tid; i < total; i += 1024)
        s += sqrtf(__uint_as_float(row_min[i])) +
             sqrtf(__uint_as_float(col_min[i]));
    red[tid] = s;
    __syncthreads();
    for (int stride = 512; stride > 0; stride >>= 1) {
        if (tid < stride) red[tid] += red[tid + stride];
        __syncthreads();
    }
    if (tid == 0) out[0] = red[0] / (float)total;
}

// ---------------------------------------------------------------------------
extern "C" void kernel_launch(void* const* d_in, const int* in_sizes, int n_in,
                              void* d_out, int out_size, void* d_ws, size_t ws_size,
                              hipStream_t stream)
{
    const float* pred   = (const float*)d_in[0];
    const float* target = (const float*)d_in[1];

    // workspace layout (512 KB total)
    unsigned* row_min = (unsigned*)d_ws;          // [4*8192] u32
    unsigned* col_min = row_min + 4 * NPTS;       // [4*8192] u32
    float*    x2      = (float*)(col_min + 4 * NPTS);  // [4*8192] f32
    float*    y2      = x2 + 4 * NPTS;                 // [4*8192] f32

    chamfer_init<<<(4 * NPTS + 255) / 256, 256, 0, stream>>>(
        pred, target, row_min, col_min, x2, y2);

    dim3 grid(NPTS / 128, NPTS / 128, 4);   // (mb, nb, b) = (64, 64, 4)
    chamfer_tiles<<<grid, 256, 0, stream>>>(
        pred, target, row_min, col_min, x2, y2);

    chamfer_final<<<1, 1024, 0, stream>>>(row_min, col_min, (float*)d_out);
}